// ConditionalLogisticRegression_32366873542997
// MI455X (gfx1250) — compile-verified
//
#include <hip/hip_runtime.h>
#include <hip/hip_bf16.h>

// Problem constants (match the reference).
#define NROWS      4194304
#define DFEAT      64
#define NSEG       262144
#define ROWS_PER_WAVE 16
#define WAVES_PER_BLOCK 8   // 256 threads, wave32
#define ROWS_PER_BLOCK (ROWS_PER_WAVE * WAVES_PER_BLOCK)  // 128

typedef __attribute__((ext_vector_type(2))) float v2f;
typedef __attribute__((ext_vector_type(8))) float v8f;

// ---------------------------------------------------------------------------
// Kernel 0: zero the per-segment accumulator (lives in d_ws).
// ---------------------------------------------------------------------------
__global__ void __launch_bounds__(256)
zero_segsum_kernel(float* __restrict__ segsum, int n) {
    int i = blockIdx.x * blockDim.x + threadIdx.x;
    if (i < n) segsum[i] = 0.0f;
}

// ---------------------------------------------------------------------------
// Kernel 1: y_hat[i] = exp(dot(x[i,:], W) + b) via V_WMMA_F32_16X16X4_F32,
// plus atomic accumulation of per-segment sums.
//
// One wave -> 16 rows. K=64 is consumed as 16 WMMA steps of K=4.
// A-matrix 16x4 f32 layout (ISA 7.12.2): lanes 0-15 hold M=0..15 with
// VGPR0=K0, VGPR1=K1; lanes 16-31 hold the same rows with VGPR0=K2, VGPR1=K3.
// So lane L loads float2 at x[row0 + (L&15)][k0 + 2*(L>>4)].
// B-matrix 4x16: broadcast W[k] across all 16 columns using the mirrored
// half-wave K-split, so every column of D equals the row dot-product.
// ---------------------------------------------------------------------------
__global__ void __launch_bounds__(256)
clr_pass_a_kernel(const float* __restrict__ x,
                  const float* __restrict__ W,
                  const float* __restrict__ bias_p,
                  const int*   __restrict__ sid,
                  float* __restrict__ yhat,
                  float* __restrict__ segsum) {
    const int lane   = threadIdx.x & 31;
    const int wave   = blockIdx.x * WAVES_PER_BLOCK + (threadIdx.x >> 5);
    const long long row0 = (long long)wave * ROWS_PER_WAVE;

    const int m      = lane & 15;         // row within the 16-row tile
    const int khalf  = (lane >> 4) << 1;  // 0 for lanes 0-15, 2 for lanes 16-31

    const float* __restrict__ xrow = x + (row0 + m) * DFEAT + khalf;

    v8f c = {};  // 16x16 f32 accumulator, 8 VGPRs
#pragma unroll
    for (int s = 0; s < 16; ++s) {
        const int k0 = s * 4;
        v2f a = *(const v2f*)(xrow + k0);           // A fragment (x tile)
        v2f b = *(const v2f*)(W + k0 + khalf);      // B fragment (W broadcast)
        // 8 args: (neg_a, A, neg_b, B, c_mod, C, reuse_a, reuse_b)
        c = __builtin_amdgcn_wmma_f32_16x16x4_f32(
                false, a, false, b, (short)0, c, false, false);
    }

    const float bias = bias_p[0];

    // D layout: lanes 0-15 column N=lane, VGPR r = row r; lanes 16-31 column
    // N=lane-16, VGPR r = row r+8. All columns identical, so lane 0 holds
    // rows 0-7 and lane 16 holds rows 8-15 of this tile in c[0..7].
    if (m == 0) {
        const long long base = row0 + (long long)((lane >> 4) * 8);
#pragma unroll
        for (int r = 0; r < 8; ++r) {
            const float y = __expf(c[r] + bias);
            const long long row = base + r;
            yhat[row] = y;
            // Hardware global_atomic_add_f32 into the 1MB L2-resident buffer.
            unsafeAtomicAdd(&segsum[sid[row]], y);
        }
    }
}

// ---------------------------------------------------------------------------
// Kernel 2: out[i] = yhat[i] / segsum[sid[i]]  (streaming; segsum in L2)
// ---------------------------------------------------------------------------
__global__ void __launch_bounds__(256)
clr_pass_b_kernel(const float* __restrict__ yhat,
                  const int*   __restrict__ sid,
                  const float* __restrict__ segsum,
                  float* __restrict__ out, int n) {
    int i = blockIdx.x * blockDim.x + threadIdx.x;
    if (i < n) {
        out[i] = yhat[i] / segsum[sid[i]];
    }
}

extern "C" void kernel_launch(void* const* d_in, const int* in_sizes, int n_in,
                              void* d_out, int out_size, void* d_ws, size_t ws_size,
                              hipStream_t stream) {
    const float* x    = (const float*)d_in[0];  // [N, 64] f32
    const float* W    = (const float*)d_in[1];  // [64, 1] f32
    const float* b    = (const float*)d_in[2];  // [1]     f32
    const int*   sid  = (const int*)  d_in[3];  // [N]     i32 (sorted)
    float*       out  = (float*)d_out;          // [N, 1]  f32

    // Workspace layout: yhat (N floats, 16MB) | segsum (NSEG floats, 1MB)
    float* yhat   = (float*)d_ws;
    float* segsum = yhat + NROWS;

    zero_segsum_kernel<<<(NSEG + 255) / 256, 256, 0, stream>>>(segsum, NSEG);

    clr_pass_a_kernel<<<NROWS / ROWS_PER_BLOCK, 256, 0, stream>>>(
        x, W, b, sid, yhat, segsum);

    clr_pass_b_kernel<<<(NROWS + 255) / 256, 256, 0, stream>>>(
        yhat, sid, segsum, out, NROWS);
}